// CubeSimulator_GPT_53171695125433
// MI455X (gfx1250) — compile-verified
//
#include <hip/hip_runtime.h>
#include <hip/hip_bf16.h>
#include <math.h>

// ---- problem constants (match reference) ----
#define G        96     // UP_GAL
#define NV       64     // UP_VEL
#define GAL_RES  64
#define VEL_RES  32
#define VMAXC    200.0f
#define R_TURN   2.0f
#define R_D      3.0f
#define H_Z      0.5f

#define LOG2E    1.4426950408889634f

typedef __attribute__((ext_vector_type(2))) float v2f;
typedef __attribute__((ext_vector_type(8))) float v8f;

// raw v_exp_f32 (exp2); argument here is always <= 0 so no overflow path needed,
// underflow flushes to 0 which is the exact desired Gaussian-tail behavior.
__device__ __forceinline__ float fast_exp2(float x) {
    return __builtin_amdgcn_exp2f(x);
}

// ============================================================
// Kernel 1: one wave32 per sky pixel (i,j).
//  Phase A: lanes cooperatively compute vlos[z], intensity[z] for z=0..95 -> LDS
//  Phase B: KDE contraction cube[v] = sum_z exp(-(vel[v]-vlos[z])^2/s2) * I[z]
//           via V_WMMA_F32_16X16X4_F32 chains (full f32 precision accumulate).
// ============================================================
__global__ __launch_bounds__(32)
void kde_cube_kernel(const float* __restrict__ inc_p,
                     const float* __restrict__ rot_p,
                     const float* __restrict__ lb_p,
                     const float* __restrict__ Xg,
                     const float* __restrict__ Yg,
                     const float* __restrict__ Zg,
                     const float* __restrict__ vel,
                     float* __restrict__ cube /* (G,G,NV) */)
{
    __shared__ float su[G];   // vlos per z
    __shared__ float sw[G];   // intensity per z

    const int pix  = blockIdx.x;          // 0 .. G*G-1
    const int lane = threadIdx.x;         // 0 .. 31 (wave32)

    // rotation matrix R = Rx(inc) @ Rz(rot)
    const float inc = inc_p[0];
    const float rot = rot_p[0];
    const float lb  = lb_p[0];
    const float ci = cosf(inc), si = sinf(inc);
    const float cr = cosf(rot), sr = sinf(rot);
    // R rows:
    // [ cr,    -sr,    0  ]
    // [ ci*sr,  ci*cr, -si]
    // [ si*sr,  si*cr,  ci]
    const float r00 = cr,      r01 = -sr,     r02 = 0.0f;
    const float r10 = ci * sr, r11 = ci * cr, r12 = -si;
    const float r20 = si * sr, r21 = si * cr, r22 = ci;
    // RT[2,0] = R[0,2], RT[2,1] = R[1,2]  (v_z component of DoRotationT, v_z_in = 0)
    const float p0 = r02, p1 = r12;

    const long long base = (long long)pix * G;   // (i*G + j)*G
    for (int z = lane; z < G; z += 32) {
        const long long idx = base + z;
        const float X = Xg[idx], Y = Yg[idx], Z = Zg[idx];
        const float rx = r00 * X + r01 * Y + r02 * Z;
        const float ry = r10 * X + r11 * Y + r12 * Z;
        const float rz = r20 * X + r21 * Y + r22 * Z;
        const float r  = sqrtf(rx * rx + ry * ry);
        const float vabs = VMAXC * tanhf(r / R_TURN);
        const float th   = atan2f(ry, rx);
        const float vx   = -vabs * sinf(th);
        const float vy   =  vabs * cosf(th);
        su[z] = p0 * vx + p1 * vy;                              // line-of-sight velocity
        // exp(-r/R_D - |rz|/H_Z) = exp2(log2e * (...)), arg <= 0
        sw[z] = fast_exp2((-r * (1.0f / R_D) - fabsf(rz) * (1.0f / H_Z)) * LOG2E);
    }
    __syncthreads();

    const float s2     = lb * lb;
    const float kexp   = -LOG2E / s2;     // exp(-d^2/s2) == exp2(kexp * d^2)
    const float normf  = 1.0f / sqrtf(2.0f * 3.14159265358979323846f * s2);
    const int   zoff   = (lane >= 16) ? 2 : 0;
    const int   mrow   = lane & 15;

    // 4 tiles of 16 velocities; 24 K-chunks of 4 z-samples each.
    for (int vt = 0; vt < 4; ++vt) {
        const float velv = vel[vt * 16 + mrow];
        v8f acc = {};
        #pragma unroll 4
        for (int zb = 0; zb < G; zb += 4) {
            const float u0 = su[zb + zoff];
            const float u1 = su[zb + zoff + 1];
            const float d0 = velv - u0;
            const float d1 = velv - u1;
            v2f a, b;
            a.x = fast_exp2(d0 * d0 * kexp);  // A[m=mrow][k=zoff+0]  (v_mul + v_exp_f32)
            a.y = fast_exp2(d1 * d1 * kexp);  // A[m=mrow][k=zoff+1]
            b.x = sw[zb + zoff];              // B[k=zoff+0][n] broadcast over n
            b.y = sw[zb + zoff + 1];          // B[k=zoff+1][n] broadcast over n
            acc = __builtin_amdgcn_wmma_f32_16x16x4_f32(
                      /*neg_a=*/false, a, /*neg_b=*/false, b,
                      /*c_mod=*/(short)0, acc,
                      /*reuse_a=*/false, /*reuse_b=*/false);
        }
        // D layout: lanes 0-15 hold M=0..7 in acc[0..7]; lanes 16-31 hold M=8..15.
        float* dst = cube + (long long)pix * NV + vt * 16;
        if (lane == 0) {
            #pragma unroll
            for (int m = 0; m < 8; ++m) dst[m] = acc[m] * normf;
        } else if (lane == 16) {
            #pragma unroll
            for (int m = 0; m < 8; ++m) dst[8 + m] = acc[m] * normf;
        }
    }
}

// ============================================================
// Kernel 2: separable trilinear downsample (G,G,NV) -> (VEL_RES,GAL_RES,GAL_RES)
// align_corners=False; axis2 (64->64) is identity.
// ============================================================
__global__ __launch_bounds__(256)
void downsample_kernel(const float* __restrict__ cube,
                       float* __restrict__ out)
{
    const int tid = blockIdx.x * blockDim.x + threadIdx.x;
    const int total = VEL_RES * GAL_RES * GAL_RES;
    if (tid >= total) return;

    const int v = tid % GAL_RES;
    const int b = (tid / GAL_RES) % GAL_RES;
    const int a = tid / (GAL_RES * GAL_RES);

    // axis 0: 96 -> 32, scale = 3
    float c0 = (a + 0.5f) * (float)G / (float)VEL_RES - 0.5f;
    c0 = fminf(fmaxf(c0, 0.0f), (float)(G - 1));
    const int   i0 = (int)floorf(c0);
    const int   i1 = min(i0 + 1, G - 1);
    const float w0 = c0 - (float)i0;

    // axis 1: 96 -> 64, scale = 1.5
    float c1 = (b + 0.5f) * (float)G / (float)GAL_RES - 0.5f;
    c1 = fminf(fmaxf(c1, 0.0f), (float)(G - 1));
    const int   j0 = (int)floorf(c1);
    const int   j1 = min(j0 + 1, G - 1);
    const float w1 = c1 - (float)j0;

    const long long s_i0 = (long long)i0 * G * NV;
    const long long s_i1 = (long long)i1 * G * NV;
    const float v00 = cube[s_i0 + (long long)j0 * NV + v];
    const float v01 = cube[s_i0 + (long long)j1 * NV + v];
    const float v10 = cube[s_i1 + (long long)j0 * NV + v];
    const float v11 = cube[s_i1 + (long long)j1 * NV + v];

    out[tid] = (1.0f - w0) * ((1.0f - w1) * v00 + w1 * v01)
             +         w0  * ((1.0f - w1) * v10 + w1 * v11);
}

// ============================================================
// launch
// ============================================================
extern "C" void kernel_launch(void* const* d_in, const int* in_sizes, int n_in,
                              void* d_out, int out_size, void* d_ws, size_t ws_size,
                              hipStream_t stream)
{
    (void)in_sizes; (void)n_in; (void)out_size; (void)ws_size;
    const float* inc = (const float*)d_in[0];
    const float* rot = (const float*)d_in[1];
    const float* lb  = (const float*)d_in[2];
    const float* Xg  = (const float*)d_in[3];
    const float* Yg  = (const float*)d_in[4];
    const float* Zg  = (const float*)d_in[5];
    const float* vel = (const float*)d_in[6];
    float* out  = (float*)d_out;
    float* cube = (float*)d_ws;              // G*G*NV floats = 2.25 MB scratch

    kde_cube_kernel<<<G * G, 32, 0, stream>>>(inc, rot, lb, Xg, Yg, Zg, vel, cube);

    const int total = VEL_RES * GAL_RES * GAL_RES;
    downsample_kernel<<<(total + 255) / 256, 256, 0, stream>>>(cube, out);
}